// SparseGraphAttention_22763326669309
// MI455X (gfx1250) — compile-verified
//
#include <hip/hip_runtime.h>
#include <math.h>
#include <stdint.h>

#define GN 8192
#define GF 256
#define GALPHA 0.2f

typedef __attribute__((ext_vector_type(2))) float v2f;
typedef __attribute__((ext_vector_type(8))) float v8f;
typedef __attribute__((ext_vector_type(4))) int  v4i;

typedef __attribute__((address_space(1))) v4i as1_v4i;
typedef __attribute__((address_space(3))) v4i as3_v4i;

__device__ __forceinline__ float warp_reduce_sum(float v) {
#pragma unroll
  for (int off = 16; off > 0; off >>= 1) v += __shfl_xor(v, off, 32);
  return v;
}

// ---------------------------------------------------------------------------
// Kernel 1: h = x @ W^T  (16-row tiles, 8 waves x two 16x16 f32-WMMA tiles)
// ---------------------------------------------------------------------------
__global__ void k_hgemm(const float* __restrict__ x, const float* __restrict__ W,
                        float* __restrict__ h) {
  const int i0   = blockIdx.x * 16;
  const int lane = threadIdx.x & 31;
  const int wave = threadIdx.x >> 5;
  const int r    = lane & 15;              // A row / B col within tile
  const int khi  = (lane >> 4) << 1;       // K-pair select per ISA A layout
  const int nb0  = wave * 32;
  const int nb1  = nb0 + 16;

  v8f c0 = {}; v8f c1 = {};
  const float* xrow = x + (size_t)(i0 + r) * GF;
  const float* w0   = W + (size_t)(nb0 + r) * GF;
  const float* w1   = W + (size_t)(nb1 + r) * GF;

  for (int k = 0; k < GF; k += 4) {
    const int kb = k + khi;
    v2f a;  a.x  = xrow[kb]; a.y  = xrow[kb + 1];
    v2f b0; b0.x = w0[kb];   b0.y = w0[kb + 1];     // B[kk][n] = W[n][k+kk]
    v2f b1; b1.x = w1[kb];   b1.y = w1[kb + 1];
    c0 = __builtin_amdgcn_wmma_f32_16x16x4_f32(false, a, false, b0, (short)0, c0, false, false);
    c1 = __builtin_amdgcn_wmma_f32_16x16x4_f32(false, a, false, b1, (short)0, c1, false, false);
  }

  const int mbase = i0 + ((lane >> 4) << 3);   // lanes 16-31 hold M = r+8
  const int col   = lane & 15;
#pragma unroll
  for (int rr = 0; rr < 8; ++rr) {
    h[(size_t)(mbase + rr) * GF + nb0 + col] = c0[rr];
    h[(size_t)(mbase + rr) * GF + nb1 + col] = c1[rr];
  }
}

// ---------------------------------------------------------------------------
// Kernel 2: a_src[i] = h[i].attn_src ; a_dst[i] = h[i].attn_dst (wave per row)
// ---------------------------------------------------------------------------
__global__ void k_attn(const float* __restrict__ h, const float* __restrict__ as,
                       const float* __restrict__ ad, float* __restrict__ a_src,
                       float* __restrict__ a_dst) {
  const int lane = threadIdx.x & 31;
  const int wave = threadIdx.x >> 5;
  const int row  = blockIdx.x * 8 + wave;
  const float* hr = h + (size_t)row * GF;
  float s = 0.f, d = 0.f;
#pragma unroll
  for (int t = 0; t < GF / 32; ++t) {
    const int idx = lane + 32 * t;
    const float v = hr[idx];
    s += v * as[idx];
    d += v * ad[idx];
  }
  s = warp_reduce_sum(s);
  d = warp_reduce_sum(d);
  if (lane == 0) { a_src[row] = s; a_dst[row] = d; }
}

// ---------------------------------------------------------------------------
// Kernel 3: per-row softmax stats. LeakyReLU is monotone =>
//   rowmax = max(0, leaky(a_dst[i] + max_edges a_src[j]))
// then denom = sum_edges exp(leaky(a_src[j]+a_dst[i]) - rowmax)
// ---------------------------------------------------------------------------
__global__ void k_rowstats(const float* __restrict__ adj, const float* __restrict__ a_src,
                           const float* __restrict__ a_dst, float* __restrict__ rowmax,
                           float* __restrict__ inv_denom) {
  __shared__ float red[256];
  const int i = blockIdx.x;
  const int t = threadIdx.x;
  const float di = a_dst[i];
  const float* arow = adj + (size_t)i * GN;

  float m = -INFINITY;
  for (int j = t; j < GN; j += 256) {
    if (j + 2048 < GN) __builtin_prefetch(&arow[j + 2048], 0, 0);
    if (arow[j] > 0.f) m = fmaxf(m, a_src[j]);
  }
  red[t] = m; __syncthreads();
  for (int off = 128; off > 0; off >>= 1) {
    if (t < off) red[t] = fmaxf(red[t], red[t + off]);
    __syncthreads();
  }
  const float sc   = di + red[0];              // -inf if row has no edges
  const float rmax = fmaxf(0.f, sc > 0.f ? sc : GALPHA * sc);
  __syncthreads();

  float s = 0.f;
  for (int j = t; j < GN; j += 256) {
    if (arow[j] > 0.f) {
      float v = a_src[j] + di;
      v = v > 0.f ? v : GALPHA * v;
      s += __expf(v - rmax);
    }
  }
  red[t] = s; __syncthreads();
  for (int off = 128; off > 0; off >>= 1) {
    if (t < off) red[t] += red[t + off];
    __syncthreads();
  }
  if (t == 0) {
    rowmax[i]    = rmax;
    inv_denom[i] = 1.f / (red[0] + 1e-8f);
  }
}

// ---------------------------------------------------------------------------
// Kernel 4: out = elu((e @ h) * inv_denom).
// adj is double-buffered into LDS with global_load_async_to_lds_b128 so the
// 256 MB HBM stream overlaps the WMMA k-loop; e-tiles built from LDS.
// ---------------------------------------------------------------------------
__global__ void k_aggregate(const float* __restrict__ adj, const float* __restrict__ h,
                            const float* __restrict__ a_src, const float* __restrict__ a_dst,
                            const float* __restrict__ rowmax, const float* __restrict__ inv_denom,
                            float* __restrict__ out) {
  __shared__ float adj_buf[2][16 * 64];          // double-buffered adj chunk
  __shared__ float e_lds[16 * 64];               // attention-weight tile
  __shared__ float row_dst[16], row_max[16], row_inv[16];

  const int i0   = blockIdx.x * 16;
  const int t    = threadIdx.x;
  const int lane = t & 31;
  const int wave = t >> 5;
  if (t < 16) {
    row_dst[t] = a_dst[i0 + t];
    row_max[t] = rowmax[i0 + t];
    row_inv[t] = inv_denom[i0 + t];
  }

  // Each thread owns one 16-byte segment of the 16x64 chunk.
  const int cr   = t >> 4;                       // chunk row 0..15
  const int cseg = (t & 15) << 2;                // float offset 0..60
  const float* gsrc = adj + (size_t)(i0 + cr) * GN + cseg;

  // Preload chunk 0 asynchronously.
  __builtin_amdgcn_global_load_async_to_lds_b128(
      (as1_v4i*)(uintptr_t)gsrc,
      (as3_v4i*)&adj_buf[0][cr * 64 + cseg], 0, 0);
  __builtin_amdgcn_s_wait_asynccnt(0);
  __syncthreads();

  const int nb0  = wave * 32, nb1 = nb0 + 16;
  const int ar   = lane & 15;            // A row
  const int khi  = (lane >> 4) << 1;     // A K-pair select
  const int ncol = lane & 15;            // B/D column
  v8f c0 = {}; v8f c1 = {};

  int cur = 0;
  for (int j0 = 0; j0 < GN; j0 += 64) {
    // Build the 16x64 e-tile from the LDS-staged adj chunk.
#pragma unroll
    for (int u = 0; u < 4; ++u) {
      const int idx = t + 256 * u;       // 0..1023
      const int r   = idx >> 6;          // row in tile
      const int jj  = idx & 63;          // col in chunk
      const float a = adj_buf[cur][r * 64 + jj];
      float e = 0.f;
      if (a > 0.f) {
        float v = a_src[j0 + jj] + row_dst[r];
        v = v > 0.f ? v : GALPHA * v;
        e = __expf(v - row_max[r]);
      }
      e_lds[r * 64 + jj] = e;
    }
    // Kick off the async copy of the next chunk; it completes under the WMMAs.
    if (j0 + 64 < GN)
      __builtin_amdgcn_global_load_async_to_lds_b128(
          (as1_v4i*)(uintptr_t)(gsrc + j0 + 64),
          (as3_v4i*)&adj_buf[cur ^ 1][cr * 64 + cseg], 0, 0);
    __syncthreads();                      // e_lds ready for all waves

#pragma unroll 4
    for (int kk = 0; kk < 16; ++kk) {
      const int kb = kk * 4 + khi;
      v2f a;  a.x  = e_lds[ar * 64 + kb]; a.y = e_lds[ar * 64 + kb + 1];
      const size_t jr = (size_t)(j0 + kb);
      v2f b0; b0.x = h[jr * GF + nb0 + ncol]; b0.y = h[(jr + 1) * GF + nb0 + ncol];
      v2f b1; b1.x = h[jr * GF + nb1 + ncol]; b1.y = h[(jr + 1) * GF + nb1 + ncol];
      c0 = __builtin_amdgcn_wmma_f32_16x16x4_f32(false, a, false, b0, (short)0, c0, false, false);
      c1 = __builtin_amdgcn_wmma_f32_16x16x4_f32(false, a, false, b1, (short)0, c1, false, false);
    }

    __builtin_amdgcn_s_wait_asynccnt(0);  // own async writes landed in LDS
    __syncthreads();                      // all waves' chunks visible
    cur ^= 1;
  }

  const int mhalf = (lane >> 4) << 3;
#pragma unroll
  for (int rr = 0; rr < 8; ++rr) {
    const int rl  = mhalf + rr;
    const int row = i0 + rl;
    const float inv = row_inv[rl];
    float v0 = c0[rr] * inv;
    float v1 = c1[rr] * inv;
    v0 = v0 > 0.f ? v0 : expm1f(v0);   // ELU (alpha = 1)
    v1 = v1 > 0.f ? v1 : expm1f(v1);
    out[(size_t)row * GF + nb0 + ncol] = v0;
    out[(size_t)row * GF + nb1 + ncol] = v1;
  }
}

extern "C" void kernel_launch(void* const* d_in, const int* in_sizes, int n_in,
                              void* d_out, int out_size, void* d_ws, size_t ws_size,
                              hipStream_t stream) {
  const float* x   = (const float*)d_in[0];
  const float* adj = (const float*)d_in[1];
  const float* W   = (const float*)d_in[2];
  const float* as  = (const float*)d_in[3];
  const float* ad  = (const float*)d_in[4];
  float* out = (float*)d_out;

  float* ws        = (float*)d_ws;
  float* h         = ws;                         // GN*GF floats (8 MB)
  float* a_src     = h + (size_t)GN * GF;        // GN
  float* a_dst     = a_src + GN;                 // GN
  float* rowmax    = a_dst + GN;                 // GN
  float* inv_denom = rowmax + GN;                // GN

  k_hgemm    <<<GN / 16, 256, 0, stream>>>(x, W, h);
  k_attn     <<<GN / 8,  256, 0, stream>>>(h, as, ad, a_src, a_dst);
  k_rowstats <<<GN,      256, 0, stream>>>(adj, a_src, a_dst, rowmax, inv_denom);
  k_aggregate<<<GN / 16, 256, 0, stream>>>(adj, h, a_src, a_dst, rowmax, inv_denom, out);
}